// ASAP_58033598104012
// MI455X (gfx1250) — compile-verified
//
#include <hip/hip_runtime.h>
#include <math.h>

#define N_NODES    100000
#define N_EDGES    3200000
#define NGRAPHS    256
#define NCLASSES   10
#define BN_EPS     1e-5f

typedef __attribute__((ext_vector_type(2))) float v2f;
typedef __attribute__((ext_vector_type(8))) float v8f;

// ---------------------------------------------------------------- utilities
__global__ void zero_kernel(float* __restrict__ p, int n) {
    int i = blockIdx.x * blockDim.x + threadIdx.x;
    if (i < n) p[i] = 0.0f;
}

__global__ void degree_kernel(const int* __restrict__ dst, float* __restrict__ cnt, int e) {
    int i = blockIdx.x * blockDim.x + threadIdx.x;
    if (i < e) atomicAdd(&cnt[dst[i]], 1.0f);
}

__global__ void inv_kernel(const float* __restrict__ cnt, float* __restrict__ inv, int n) {
    int i = blockIdx.x * blockDim.x + threadIdx.x;
    if (i < n) inv[i] = 1.0f / fmaxf(cnt[i], 1.0f);
}

// ------------------------------------------------- edge scatter (mean numerator)
// s[dst[e], :] += t[src[e], :]  over d channels, float4 per thread
__global__ void scatter_kernel(const float* __restrict__ t, const int* __restrict__ src,
                               const int* __restrict__ dst, float* __restrict__ s,
                               int e, int d4) {
    int tid = blockIdx.x * blockDim.x + threadIdx.x;
    int total = e * d4;
    if (tid >= total) return;
    int edge = tid / d4;
    int g    = tid - edge * d4;
    int sn = src[edge];
    int dn = dst[edge];
    const float4 v = ((const float4*)(t + (size_t)sn * d4 * 4))[g];
    float* outp = s + (size_t)dn * d4 * 4 + g * 4;
    atomicAdd(outp + 0, v.x);
    atomicAdd(outp + 1, v.y);
    atomicAdd(outp + 2, v.z);
    atomicAdd(outp + 3, v.w);
}

// ------------------------------------------------- fused dual-GEMM (fp32 WMMA)
// out[m,n] = (rs1[m] * A1[m,:]) @ W1 [+ A2 @ W2] [+ bias[n]] [+ extra_rs[m]*extra[m,n]]
// One wave per 16x16 tile; M%16==0, K%4==0, Nc%16==0.
__global__ void gemm2_wmma_kernel(const float* __restrict__ A1, const float* __restrict__ rs1,
                                  const float* __restrict__ W1,
                                  const float* __restrict__ A2, const float* __restrict__ W2,
                                  const float* __restrict__ bias,
                                  const float* __restrict__ extra, const float* __restrict__ extra_rs,
                                  float* __restrict__ out,
                                  int M, int K, int Nc) {
    int wave = (blockIdx.x * blockDim.x + threadIdx.x) >> 5;
    int lane = threadIdx.x & 31;
    int tilesN = Nc >> 4;
    int totalTiles = (M >> 4) * tilesN;
    if (wave >= totalTiles) return;           // wave-uniform: EXEC stays all-1s
    int tm = wave / tilesN;
    int tn = wave - tm * tilesN;
    int m0 = tm << 4, n0 = tn << 4;
    int half = lane >> 4;                     // 0 or 1
    int mr   = lane & 15;                     // A-row / B-col / C-col

    int rowA = m0 + mr;
    float scale1 = rs1 ? rs1[rowA] : 1.0f;

    v8f c = {};
    for (int kb = 0; kb < K; kb += 4) {
        int k0 = kb + half * 2;
        v2f a, b;
        // A 16x4: lane l%16 = M, (l/16)*2+r = K
        a[0] = A1[(size_t)rowA * K + k0]     * scale1;
        a[1] = A1[(size_t)rowA * K + k0 + 1] * scale1;
        // B 4x16: VGPR r + half*2 = K, lane%16 = N
        b[0] = W1[(size_t)k0 * Nc + n0 + mr];
        b[1] = W1[(size_t)(k0 + 1) * Nc + n0 + mr];
        c = __builtin_amdgcn_wmma_f32_16x16x4_f32(false, a, false, b, (short)0, c, false, false);
        if (A2) {
            v2f a2, b2;
            a2[0] = A2[(size_t)rowA * K + k0];
            a2[1] = A2[(size_t)rowA * K + k0 + 1];
            b2[0] = W2[(size_t)k0 * Nc + n0 + mr];
            b2[1] = W2[(size_t)(k0 + 1) * Nc + n0 + mr];
            c = __builtin_amdgcn_wmma_f32_16x16x4_f32(false, a2, false, b2, (short)0, c, false, false);
        }
    }
    // C/D layout: VGPR v -> row = v + half*8, col = lane%16
    for (int v = 0; v < 8; v++) {
        int row = m0 + v + half * 8;
        int col = n0 + mr;
        float r = c[v];
        if (bias)  r += bias[col];
        if (extra) r += extra[(size_t)row * Nc + col] * (extra_rs ? extra_rs[row] : 1.0f);
        out[(size_t)row * Nc + col] = r;
    }
}

// ------------------------------------------------- batch-norm (training stats) + ELU
__global__ void bn_stats_kernel(const float* __restrict__ h, float* __restrict__ stats,
                                int n, int C) {
    int c   = threadIdx.x % C;
    int rpb = blockDim.x / C;
    int r0  = blockIdx.x * rpb + threadIdx.x / C;
    int stride = gridDim.x * rpb;
    float s = 0.0f, sq = 0.0f;
    for (int r = r0; r < n; r += stride) {
        float v = h[(size_t)r * C + c];
        s += v; sq += v * v;
    }
    atomicAdd(&stats[c], s);
    atomicAdd(&stats[C + c], sq);
}

__global__ void bn_final_kernel(const float* __restrict__ stats,
                                const float* __restrict__ g, const float* __restrict__ be,
                                float* __restrict__ ab, int n, int C) {
    int c = threadIdx.x;
    if (c >= C) return;
    float mean = stats[c] / (float)n;
    float var  = stats[C + c] / (float)n - mean * mean;
    float a = g[c] * rsqrtf(var + BN_EPS);
    ab[c]     = a;
    ab[C + c] = be[c] - mean * a;
}

__global__ void bn_apply_elu_kernel(float* __restrict__ h, const float* __restrict__ ab,
                                    int n, int C) {
    int i = blockIdx.x * blockDim.x + threadIdx.x;
    if (i >= n * C) return;
    int c = i % C;
    float v = h[i] * ab[c] + ab[C + c];
    h[i] = v > 0.0f ? v : expm1f(v);
}

// ------------------------------------------------- global mean pool + head
__global__ void pool_scatter_kernel(const float* __restrict__ h, const int* __restrict__ batch,
                                    float* __restrict__ pooled, int n, int C) {
    int i = blockIdx.x * blockDim.x + threadIdx.x;
    if (i >= n * C) return;
    int node = i / C, c = i - node * C;
    atomicAdd(&pooled[batch[node] * C + c], h[i]);
}

__global__ void pool_count_kernel(const int* __restrict__ batch, float* __restrict__ gcnt, int n) {
    int i = blockIdx.x * blockDim.x + threadIdx.x;
    if (i < n) atomicAdd(&gcnt[batch[i]], 1.0f);
}

__global__ void pool_div_kernel(float* __restrict__ pooled, const float* __restrict__ gcnt, int C) {
    int i = blockIdx.x * blockDim.x + threadIdx.x;
    if (i >= NGRAPHS * C) return;
    pooled[i] /= fmaxf(gcnt[i / C], 1.0f);
}

// relu(zpre) @ w_lin2 + b_lin2 -> log_softmax ; one thread per graph
__global__ void head_kernel(const float* __restrict__ zpre, const float* __restrict__ w2,
                            const float* __restrict__ b2, float* __restrict__ out) {
    int g = blockIdx.x * blockDim.x + threadIdx.x;
    if (g >= NGRAPHS) return;
    float lg[NCLASSES];
    float mx = -1e30f;
    for (int j = 0; j < NCLASSES; j++) {
        float acc = b2[j];
        for (int k = 0; k < 64; k++) {
            float z = zpre[g * 64 + k];
            z = z > 0.0f ? z : 0.0f;
            acc += z * w2[k * NCLASSES + j];
        }
        lg[j] = acc;
        mx = fmaxf(mx, acc);
    }
    float se = 0.0f;
    for (int j = 0; j < NCLASSES; j++) se += expf(lg[j] - mx);
    float lse = mx + logf(se);
    for (int j = 0; j < NCLASSES; j++) out[g * NCLASSES + j] = lg[j] - lse;
}

// ---------------------------------------------------------------- driver
static inline int cdiv(int a, int b) { return (a + b - 1) / b; }

extern "C" void kernel_launch(void* const* d_in, const int* in_sizes, int n_in,
                              void* d_out, int out_size, void* d_ws, size_t ws_size,
                              hipStream_t stream) {
    const float* x     = (const float*)d_in[0];
    const int*   src   = (const int*)d_in[1];
    const int*   dst   = src + N_EDGES;
    const int*   batch = (const int*)d_in[2];

    const float* w_rel[4]  = {(const float*)d_in[3],  (const float*)d_in[8],
                              (const float*)d_in[13], (const float*)d_in[18]};
    const float* w_root[4] = {(const float*)d_in[4],  (const float*)d_in[9],
                              (const float*)d_in[14], (const float*)d_in[19]};
    const float* bb[4]     = {(const float*)d_in[5],  (const float*)d_in[10],
                              (const float*)d_in[15], (const float*)d_in[20]};
    const float* gg[4]     = {(const float*)d_in[6],  (const float*)d_in[11],
                              (const float*)d_in[16], (const float*)d_in[21]};
    const float* be[4]     = {(const float*)d_in[7],  (const float*)d_in[12],
                              (const float*)d_in[17], (const float*)d_in[22]};
    const float* w_lin1 = (const float*)d_in[23];
    const float* b_lin1 = (const float*)d_in[24];
    const float* w_lin2 = (const float*)d_in[25];
    const float* b_lin2 = (const float*)d_in[26];
    float* out = (float*)d_out;

    // workspace carve-up (floats)
    float* ws     = (float*)d_ws;
    float* cnt    = ws;                         ws += N_NODES;
    float* inv    = ws;                         ws += N_NODES;
    float* s      = ws;                         ws += (size_t)N_NODES * 64;
    float* t      = ws;                         ws += (size_t)N_NODES * 16;
    float* hA     = ws;                         ws += (size_t)N_NODES * 64;
    float* hB     = ws;                         ws += (size_t)N_NODES * 64;
    float* stats  = ws;                         ws += 128;
    float* ab     = ws;                         ws += 128;
    float* pooled = ws;                         ws += NGRAPHS * 64;
    float* gcnt   = ws;                         ws += NGRAPHS;
    float* zpre   = ws;                         ws += NGRAPHS * 64;

    const int B = 256;
    auto zero = [&](float* p, int n) {
        zero_kernel<<<cdiv(n, B), B, 0, stream>>>(p, n);
    };
    auto gemm2 = [&](const float* A1, const float* rs1, const float* W1,
                     const float* A2, const float* W2, const float* bias,
                     const float* extra, const float* extra_rs,
                     float* o, int M, int K, int Nc) {
        int waves = (M >> 4) * (Nc >> 4);
        gemm2_wmma_kernel<<<cdiv(waves * 32, B), B, 0, stream>>>(
            A1, rs1, W1, A2, W2, bias, extra, extra_rs, o, M, K, Nc);
    };
    auto bnorm = [&](float* h, const float* gamma, const float* beta, int C) {
        zero(stats, 2 * C);
        bn_stats_kernel<<<256, B, 0, stream>>>(h, stats, N_NODES, C);
        bn_final_kernel<<<1, 64, 0, stream>>>(stats, gamma, beta, ab, N_NODES, C);
        bn_apply_elu_kernel<<<cdiv(N_NODES * C, B), B, 0, stream>>>(h, ab, N_NODES, C);
    };
    auto scat = [&](const float* tin, float* sbuf, int d) {
        zero(sbuf, N_NODES * d);
        int d4 = d / 4;
        scatter_kernel<<<cdiv(N_EDGES * d4, B), B, 0, stream>>>(tin, src, dst, sbuf, N_EDGES, d4);
    };

    // in-degree (shared by all layers)
    zero(cnt, N_NODES);
    degree_kernel<<<cdiv(N_EDGES, B), B, 0, stream>>>(dst, cnt, N_EDGES);
    inv_kernel<<<cdiv(N_NODES, B), B, 0, stream>>>(cnt, inv, N_NODES);

    // ---- layer 1: 64 -> 16 (pre-multiply by w_rel: aggregate in 16-space)
    gemm2(x, nullptr, w_rel[0], nullptr, nullptr, nullptr, nullptr, nullptr, t, N_NODES, 64, 16);
    scat(t, s, 16);
    gemm2(x, nullptr, w_root[0], nullptr, nullptr, bb[0], s, inv, hA, N_NODES, 64, 16);
    bnorm(hA, gg[0], be[0], 16);

    // ---- layer 2: 16 -> 32 (aggregate in 16-space, then dual GEMM)
    scat(hA, s, 16);
    gemm2(s, inv, w_rel[1], hA, w_root[1], bb[1], nullptr, nullptr, hB, N_NODES, 16, 32);
    bnorm(hB, gg[1], be[1], 32);

    // ---- layer 3: 32 -> 32
    scat(hB, s, 32);
    gemm2(s, inv, w_rel[2], hB, w_root[2], bb[2], nullptr, nullptr, hA, N_NODES, 32, 32);
    bnorm(hA, gg[2], be[2], 32);

    // ---- layer 4: 32 -> 64
    scat(hA, s, 32);
    gemm2(s, inv, w_rel[3], hA, w_root[3], bb[3], nullptr, nullptr, hB, N_NODES, 32, 64);
    bnorm(hB, gg[3], be[3], 64);

    // ---- global mean pool
    zero(pooled, NGRAPHS * 64);
    zero(gcnt, NGRAPHS);
    pool_scatter_kernel<<<cdiv(N_NODES * 64, B), B, 0, stream>>>(hB, batch, pooled, N_NODES, 64);
    pool_count_kernel<<<cdiv(N_NODES, B), B, 0, stream>>>(batch, gcnt, N_NODES);
    pool_div_kernel<<<cdiv(NGRAPHS * 64, B), B, 0, stream>>>(pooled, gcnt, 64);

    // ---- head: lin1 (WMMA) + relu + lin2 + log_softmax
    gemm2(pooled, nullptr, w_lin1, nullptr, nullptr, b_lin1, nullptr, nullptr, zpre, NGRAPHS, 64, 64);
    head_kernel<<<cdiv(NGRAPHS, 64), 64, 0, stream>>>(zpre, w_lin2, b_lin2, out);
}